// MaxPlus_74483322847323
// MI455X (gfx1250) — compile-verified
//
#include <hip/hip_runtime.h>
#include <stdint.h>

// Max-plus (tropical) matmul: y[b,o] = max_j (x[b,j] + a[o,j]) + bias[o]
// B=1024, O=512, J=512, fp32. VALU-bound (max-reduction => WMMA inapplicable).
// CDNA5 paths used: global_load_async_to_lds_b128 double-buffered pipeline,
// s_wait_asynccnt, XOR-swizzled LDS (conflict-free ds_load_b128),
// global_atomic_max_num_f32 cross-block combine.

#define IN_F   512
#define OUT_F  512
#define TILE   64      // output tile: 64 batch rows x 64 out features
#define KC     32      // J-chunk staged in LDS per pipeline stage
#define LDST   36      // padded LDS row stride in floats (144B = 9*16B)
#define JSPLIT 2       // J-reduction split across grid.z for occupancy
#define JPART  (IN_F / JSPLIT)   // 256
#define NCHNK  (JPART / KC)      // 8

__device__ __forceinline__ uint32_t lds_u32(const void* p) {
  // Generic LDS pointers are {shared_aperture_hi32, lds_offset_lo32};
  // async-to-LDS wants the lo32 (wave-relative) LDS byte address.
  return (uint32_t)(uintptr_t)p;
}

__device__ __forceinline__ void async_copy_b128(uint32_t lds_addr, const void* gptr) {
  asm volatile("global_load_async_to_lds_b128 %0, %1, off"
               :: "v"(lds_addr), "v"((unsigned long long)(uintptr_t)gptr)
               : "memory");
}

__device__ __forceinline__ void wait_async_le4() {
  asm volatile("s_wait_asynccnt 4" ::: "memory");
}
__device__ __forceinline__ void wait_async_le0() {
  asm volatile("s_wait_asynccnt 0" ::: "memory");
}

__device__ __forceinline__ void atomic_max_f32(float* p, float v) {
  // GLOBAL_ATOMIC_MAX_NUM_F32, no-return (fire-and-forget), device scope.
  asm volatile("global_atomic_max_num_f32 %0, %1, off scope:SCOPE_DEV"
               :: "v"((unsigned long long)(uintptr_t)p), "v"(v)
               : "memory");
}

// Column swizzle: logical 16B segment s of row r stored at segment s ^ ((r>>4)&3).
// Makes the 16-distinct-row read pattern cover all 64 LDS banks exactly once.
__device__ __forceinline__ int swz(int row, int s) {
  return (s ^ ((row >> 4) & 3)) << 2;   // float column
}

__global__ void init_neg_inf(float4* __restrict__ y4, int n4) {
  int i = (int)blockIdx.x * (int)blockDim.x + (int)threadIdx.x;
  if (i < n4) {
    const float ninf = -__builtin_inff();
    float4 v; v.x = ninf; v.y = ninf; v.z = ninf; v.w = ninf;
    y4[i] = v;
  }
}

__global__ __launch_bounds__(256) void maxplus_async_kernel(
    const float* __restrict__ x,     // [B][IN_F]
    const float* __restrict__ a,     // [OUT_F][IN_F]
    const float* __restrict__ bias,  // [OUT_F]
    float* __restrict__ y,           // [B][OUT_F], pre-filled with -inf
    int B) {
  __shared__ float xs[2][TILE][LDST];
  __shared__ float as[2][TILE][LDST];

  const int tid   = (int)threadIdx.x;
  const int obase = (int)blockIdx.x * TILE;
  const int bbase = (int)blockIdx.y * TILE;
  const int jbase = (int)blockIdx.z * JPART;
  const int to    = tid & 15;   // 16 groups of 4 output features
  const int tb    = tid >> 4;   // 16 groups of 4 batch rows

  // Stage one KC-wide chunk of both tiles via async b128 copies (4 per wave).
  auto issue = [&](int jc, int buf) {
#pragma unroll
    for (int i = 0; i < 2; ++i) {
      const int t   = tid + i * 256;
      const int row = t >> 3;        // 0..63
      const int s   = t & 7;         // logical 16B segment within the row
      const int col = swz(row, s);   // swizzled float column
      async_copy_b128(lds_u32(&xs[buf][row][col]),
                      x + (size_t)(bbase + row) * IN_F + jbase + jc + s * 4);
      async_copy_b128(lds_u32(&as[buf][row][col]),
                      a + (size_t)(obase + row) * IN_F + jbase + jc + s * 4);
    }
  };

  float m[4][4];
#pragma unroll
  for (int bi = 0; bi < 4; ++bi)
#pragma unroll
    for (int oi = 0; oi < 4; ++oi)
      m[bi][oi] = -__builtin_inff();

  issue(0, 0);

  for (int c = 0; c < NCHNK; ++c) {
    const int buf = c & 1;
    if (c + 1 < NCHNK) {
      issue((c + 1) * KC, buf ^ 1);  // prefetch next chunk
      wait_async_le4();              // chunk c complete; c+1 may be in flight
    } else {
      wait_async_le0();
    }
    __syncthreads();                 // chunk-c data visible to all waves

#pragma unroll 4
    for (int g = 0; g < KC / 4; ++g) {
      float4 xv[4], av[4];
#pragma unroll
      for (int i = 0; i < 4; ++i) {
        const int r = tb * 4 + i;
        xv[i] = *reinterpret_cast<const float4*>(&xs[buf][r][swz(r, g)]);
      }
#pragma unroll
      for (int i = 0; i < 4; ++i) {
        const int r = to * 4 + i;
        av[i] = *reinterpret_cast<const float4*>(&as[buf][r][swz(r, g)]);
      }
#pragma unroll
      for (int bi = 0; bi < 4; ++bi) {
#pragma unroll
        for (int oi = 0; oi < 4; ++oi) {
          const float t0 = xv[bi].x + av[oi].x;
          const float t1 = xv[bi].y + av[oi].y;
          const float t2 = xv[bi].z + av[oi].z;
          const float t3 = xv[bi].w + av[oi].w;
          // Shaped for v_max3_num_f32 formation.
          m[bi][oi] = fmaxf(m[bi][oi], fmaxf(fmaxf(t0, t1), fmaxf(t2, t3)));
        }
      }
    }
    __syncthreads();  // all reads of buf done before it is overwritten
  }

  // Combine J-partials across grid.z with exact fp max atomics.
  // max(rnd(m1+b), rnd(m2+b)) == rnd(max(m1,m2)+b): bit-identical to reference.
  const float4 bv = *reinterpret_cast<const float4*>(bias + obase + to * 4);
#pragma unroll
  for (int bi = 0; bi < 4; ++bi) {
    float* row = y + (size_t)(bbase + tb * 4 + bi) * OUT_F + obase + to * 4;
    atomic_max_f32(row + 0, m[bi][0] + bv.x);
    atomic_max_f32(row + 1, m[bi][1] + bv.y);
    atomic_max_f32(row + 2, m[bi][2] + bv.z);
    atomic_max_f32(row + 3, m[bi][3] + bv.w);
  }
}

extern "C" void kernel_launch(void* const* d_in, const int* in_sizes, int n_in,
                              void* d_out, int out_size, void* d_ws, size_t ws_size,
                              hipStream_t stream) {
  (void)n_in; (void)d_ws; (void)ws_size; (void)out_size;
  const float* x    = (const float*)d_in[0];  // [B, 512]
  const float* a    = (const float*)d_in[1];  // [512, 512]
  const float* bias = (const float*)d_in[2];  // [512]
  float* y          = (float*)d_out;          // [B, 512]

  const int B  = in_sizes[0] / IN_F;          // 1024
  const int n4 = (B * OUT_F) / 4;

  init_neg_inf<<<(n4 + 255) / 256, 256, 0, stream>>>((float4*)y, n4);

  dim3 grid(OUT_F / TILE, B / TILE, JSPLIT);  // (8, 16, 2) = 256 blocks
  dim3 block(256);
  maxplus_async_kernel<<<grid, block, 0, stream>>>(x, a, bias, y, B);
}